// Attention_89721866814196
// MI455X (gfx1250) — compile-verified
//
#include <hip/hip_runtime.h>
#include <math.h>

typedef __attribute__((ext_vector_type(2))) float f32x2;
typedef __attribute__((ext_vector_type(8))) float f32x8;

#define HIDDEN 64
#define S_IN   2048
#define S_T    2048
#define BATCH  8

// ---- dynamic shared memory layout (in floats) ----
#define TS_STRIDE   68                         // 16 x 68 t-tile (padded vs bank conflicts)
#define TS_OFF      0
#define RED_OFF     (16*TS_STRIDE)             // 1088: 16 x 17 reduction partials
#define RED_STRIDE  17
#define STAT_OFF    (RED_OFF + 16*RED_STRIDE)  // 1360: mean[16], max[16], invsum[16]
#define SC_OFF      (STAT_OFF + 48)            // 1408 (x4B = 5632, 16B aligned)
#define SC_STRIDE   2052                       // 2048+4: conflict-free WMMA D stores
#define SMEM_FLOATS (SC_OFF + 16*SC_STRIDE)
#define SMEM_BYTES  (SMEM_FLOATS * 4)          // ~137 KB -> 2 WGs per 320KB WGP

__global__ __launch_bounds__(256) void attn_fused(
    const float* __restrict__ input_encode,    // (S_IN, B, H)
    const float* __restrict__ target_encode,   // (S_T, B, H)
    const unsigned char* __restrict__ mask,    // (B, S_T, S_IN) bool
    const float* __restrict__ Wm,              // (H, H)
    const float* __restrict__ bias,            // (H)
    float* __restrict__ out)                   // (B, S_T, S_IN)
{
    extern __shared__ float smem[];
    float* t_s  = smem + TS_OFF;
    float* red  = smem + RED_OFF;
    float* stat = smem + STAT_OFF;
    float* sc   = smem + SC_OFF;

    const int tid = threadIdx.x;
    const int bi  = blockIdx.x;
    const int b   = bi & (BATCH - 1);
    const int t0  = (bi >> 3) * 16;
    const float NEG_INF = -__builtin_inff();

    // ---------- Phase A: t_s[m][o] = sum_h target[t0+m][b][h]*W[o][h] + bias[o]
    {
        const int m  = tid >> 4;        // 0..15
        const int o0 = tid & 15;        // 0..15, covers o0, o0+16, o0+32, o0+48
        const float* tg = target_encode + ((size_t)(t0 + m) * BATCH + b) * HIDDEN;
        float a0 = bias[o0], a1 = bias[o0 + 16], a2 = bias[o0 + 32], a3 = bias[o0 + 48];
        #pragma unroll 8
        for (int h = 0; h < HIDDEN; ++h) {
            float x = tg[h];
            a0 = fmaf(x, Wm[(o0     ) * HIDDEN + h], a0);
            a1 = fmaf(x, Wm[(o0 + 16) * HIDDEN + h], a1);
            a2 = fmaf(x, Wm[(o0 + 32) * HIDDEN + h], a2);
            a3 = fmaf(x, Wm[(o0 + 48) * HIDDEN + h], a3);
        }
        t_s[m * TS_STRIDE + o0     ] = a0;
        t_s[m * TS_STRIDE + o0 + 16] = a1;
        t_s[m * TS_STRIDE + o0 + 32] = a2;
        t_s[m * TS_STRIDE + o0 + 48] = a3;
    }
    __syncthreads();

    // ---------- Phase B: scores[t-tile, s] via V_WMMA_F32_16X16X4_F32 chains
    {
        const int w    = tid >> 5;           // wave 0..7 -> s range [256w, 256w+256)
        const int lane = tid & 31;
        const int n    = lane & 15;          // M (for A) / N (for B) index
        const int h2   = (lane >> 4) * 2;    // K sub-offset: lanes 0-15 -> K 0,1; 16-31 -> K 2,3
        const int rowoff = (lane >> 4) * 8;  // D row offset per C/D VGPR layout

        // A fragment: t rows, K striped per 16x4 f32 A layout; resident for all 16 s-tiles
        f32x2 a[16];
        #pragma unroll
        for (int j = 0; j < 16; ++j)
            a[j] = *(const f32x2*)&t_s[n * TS_STRIDE + 4 * j + h2];

        const int s_wave = w * 256;
        for (int st = 0; st < 16; ++st) {
            const int s0 = s_wave + st * 16;
            const float* bp = input_encode + ((size_t)(s0 + n) * BATCH + b) * HIDDEN + h2;
            f32x2 bb[16];
            #pragma unroll
            for (int j = 0; j < 16; ++j)
                bb[j] = *(const f32x2*)(bp + 4 * j);      // global_load_b64

            f32x8 c = {0.f, 0.f, 0.f, 0.f, 0.f, 0.f, 0.f, 0.f};
            #pragma unroll
            for (int j = 0; j < 16; ++j)                  // K = 64 = 16 x 4
                c = __builtin_amdgcn_wmma_f32_16x16x4_f32(
                        false, a[j], false, bb[j], (short)0, c, false, false);

            #pragma unroll
            for (int j = 0; j < 8; ++j)                   // D: row j+rowoff, col s0+n
                sc[(j + rowoff) * SC_STRIDE + s0 + n] = c[j];
        }
    }
    __syncthreads();

    // ---------- Phase C: mean -> abs-center(+mask) -> max -> exp-sum -> normalize
    const int r  = tid >> 4;      // row 0..15 (16 threads per row)
    const int c0 = tid & 15;
    float* scr = sc + r * SC_STRIDE;

    // pass 1: row mean (mask NOT applied, matching reference order)
    {
        float s = 0.f;
        #pragma unroll 4
        for (int i = 0; i < S_IN / 16; ++i) s += scr[c0 + 16 * i];
        red[r * RED_STRIDE + c0] = s;
    }
    __syncthreads();
    if (tid < 16) {
        float s = 0.f;
        #pragma unroll
        for (int i = 0; i < 16; ++i) s += red[tid * RED_STRIDE + i];
        stat[tid] = s * (1.0f / (float)S_IN);
    }
    __syncthreads();

    // pass 2: v = |x - mean|; masked -> -inf; row max
    {
        const float mu = stat[r];
        const unsigned char* mrow = mask + ((size_t)b * S_T + (t0 + r)) * (size_t)S_IN;
        float mx = NEG_INF;
        for (int i = 0; i < S_IN / 16; ++i) {
            int c = c0 + 16 * i;
            float v = fabsf(scr[c] - mu);
            if (mrow[c]) v = NEG_INF;
            scr[c] = v;
            mx = fmaxf(mx, v);
        }
        red[r * RED_STRIDE + c0] = mx;
    }
    __syncthreads();
    if (tid < 16) {
        float mx = NEG_INF;
        #pragma unroll
        for (int i = 0; i < 16; ++i) mx = fmaxf(mx, red[tid * RED_STRIDE + i]);
        stat[16 + tid] = mx;
    }
    __syncthreads();

    // pass 3: e = exp(v - max); row sum
    {
        const float mx = stat[16 + r];
        float s = 0.f;
        for (int i = 0; i < S_IN / 16; ++i) {
            int c = c0 + 16 * i;
            float e = __expf(scr[c] - mx);
            scr[c] = e;
            s += e;
        }
        red[r * RED_STRIDE + c0] = s;
    }
    __syncthreads();
    if (tid < 16) {
        float s = 0.f;
        #pragma unroll
        for (int i = 0; i < 16; ++i) s += red[tid * RED_STRIDE + i];
        stat[32 + tid] = 1.0f / s;
    }
    __syncthreads();

    // pass 4: coalesced float4 stores of normalized probabilities
    {
        float* orow = out + ((size_t)b * S_T + t0) * (size_t)S_IN;
        for (int i4 = tid; i4 < (16 * S_IN) / 4; i4 += 256) {
            int e  = i4 * 4;
            int rr = e >> 11;            // /2048
            int cc = e & (S_IN - 1);
            float inv = stat[32 + rr];
            const float* p = sc + rr * SC_STRIDE + cc;
            float4 v = make_float4(p[0] * inv, p[1] * inv, p[2] * inv, p[3] * inv);
            *(float4*)&orow[(size_t)rr * S_IN + cc] = v;
        }
    }
}

extern "C" void kernel_launch(void* const* d_in, const int* in_sizes, int n_in,
                              void* d_out, int out_size, void* d_ws, size_t ws_size,
                              hipStream_t stream)
{
    (void)in_sizes; (void)n_in; (void)out_size; (void)d_ws; (void)ws_size;

    const float*         input_encode  = (const float*)d_in[0];
    const float*         target_encode = (const float*)d_in[1];
    const unsigned char* mask          = (const unsigned char*)d_in[2];
    const float*         Wm            = (const float*)d_in[3];
    const float*         bias          = (const float*)d_in[4];
    float*               out           = (float*)d_out;

    // ~137 KB dynamic LDS (> default 64KB cap) — CDNA5 WGP has 320 KB
    hipFuncSetAttribute((const void*)attn_fused,
                        hipFuncAttributeMaxDynamicSharedMemorySize, SMEM_BYTES);

    dim3 grid(BATCH * (S_T / 16));   // 1024 workgroups: (b, 16-row t-tile)
    dim3 block(256);                 // 8 wave32s; each wave owns 256 s-columns
    attn_fused<<<grid, block, SMEM_BYTES, stream>>>(
        input_encode, target_encode, mask, Wm, bias, out);
}